// KnowledgeAttention_73495480369367
// MI455X (gfx1250) — compile-verified
//
#include <hip/hip_runtime.h>
#include <hip/hip_bf16.h>
#include <math.h>

// Problem constants (from reference)
#define B_  8
#define T_  128
#define K_  256
#define D_  512     // SRC = TGT = KNOW
#define F_  1536    // concat feature dim = KNOW + TGT + SRC

// GEMM tiling
#define BM  128     // block M tile (8 waves x 16 rows)
#define BN  64      // block N tile (4 WMMA tiles per wave)
#define BK  64      // K slice staged in LDS per iteration
#define PAD 4       // LDS row pad (dwords): 16B-aligned rows, conflict-free frags
#define LDW (BK + PAD)

typedef __attribute__((ext_vector_type(2))) float v2f;
typedef __attribute__((ext_vector_type(8))) float v8f;
typedef __attribute__((ext_vector_type(4))) unsigned int u32x4;
typedef __attribute__((ext_vector_type(8))) int i32x8;
typedef __attribute__((ext_vector_type(4))) int i32x4;

// ---------------------------------------------------------------------------
// LDS-tiled WMMA f32 GEMM:  Y[m,n] = sum_k X[m,k] * W[n,k] (+ bias[n])
//
// Block: 256 threads = 8 waves, computing a 128x64 output tile.
//   - W tile (64 rows x 64 k) is DMA'd into LDS by the Tensor Data Mover
//     (tensor_load_to_lds, TENSORcnt), with TDM pad_enable inserting a
//     4-dword pad per 64-dword row -> LDS row stride 68 dwords (bank-safe,
//     16B aligned).
//   - X tile (128 x 64) is staged by coalesced float4 cooperative loads.
//   - Each wave owns 16 rows and all 64 cols: 4 f32 accumulators, inner loop
//     is 5 ds_load_b64 + 4 back-to-back V_WMMA_F32_16X16X4_F32 per K-step.
// A (16x4 f32) lane layout per ISA 7.12.2: lanes 0-15 -> M=lane, VGPR0=K0,
// VGPR1=K1; lanes 16-31 -> K2/K3. B (4x16) mirrored with N across lanes.
// ---------------------------------------------------------------------------
__global__ void wmma_gemm_xwt(const float* __restrict__ X,
                              const float* __restrict__ W,
                              const float* __restrict__ bias,
                              float* __restrict__ Y,
                              int M, int N, int Kd, int Tdim, int store_tb)
{
  __shared__ float s_X[BM][LDW];
  __shared__ float s_W[BN][LDW];

  const int tilesN = N / BN;
  const int bm = (int)blockIdx.x / tilesN;
  const int bn = (int)blockIdx.x - bm * tilesN;
  const int m0 = bm * BM;
  const int n0 = bn * BN;

  const int tid  = (int)threadIdx.x;
  const int w    = tid >> 5;          // wave 0..7
  const int lane = tid & 31;
  const int half = lane >> 4;         // 0 -> K pair {0,1}, 1 -> {2,3}
  const int l    = lane & 15;

  v8f acc0 = {}, acc1 = {}, acc2 = {}, acc3 = {};

  for (int k0 = 0; k0 < Kd; k0 += BK) {
    // ---- stage X tile: coalesced float4 loads, 128x64 -> padded LDS ----
    for (int i = tid * 4; i < BM * BK; i += 256 * 4) {
      const int r = i / BK;
      const int c = i - r * BK;
      const float4 t = *(const float4*)(X + (size_t)(m0 + r) * Kd + k0 + c);
      *(float4*)&s_X[r][c] = t;
    }

    // ---- stage W tile via Tensor Data Mover (wave 0 issues, all wait) ----
    if (tid < 32) {
      const unsigned long long gaddr =
          (unsigned long long)(const void*)(W + (size_t)n0 * Kd + k0);
      const unsigned lds_off = (unsigned)(size_t)&s_W[0][0]; // LDS byte offset

      u32x4 g0;
      g0[0] = 1u;                                     // count=1 (valid D#)
      g0[1] = lds_off;                                // lds_addr
      g0[2] = (unsigned)(gaddr & 0xFFFFFFFFull);      // global_addr[31:0]
      g0[3] = (unsigned)((gaddr >> 32) & 0x01FFFFFFull) | (2u << 30); // type=2

      i32x8 g1;
      g1[0] = (int)((2u << 16)        // data_size = 4B
                  | (1u << 20)        // pad_enable
                  | (5u << 22)        // pad_interval: 2^(5+1)=64 dwords
                  | (3u << 25));      // pad_amount: 3+1 = 4 dwords
      g1[1] = (int)(((unsigned)Kd & 0xFFFFu) << 16);           // tensor_dim0 lo
      g1[2] = (int)((((unsigned)Kd >> 16) & 0xFFFFu)           // tensor_dim0 hi
                  | ((unsigned)BN << 16));                     // tensor_dim1 lo
      g1[3] = (int)((unsigned)BK << 16);                       // tile_dim0
      g1[4] = (int)(unsigned)BN;                               // tile_dim1
      g1[5] = (int)(unsigned)Kd;                               // dim0_stride lo
      g1[6] = 0;                                               // strides hi
      g1[7] = 0;

      const i32x4 gz4 = {0, 0, 0, 0};                          // 2D: groups 2/3
      const i32x8 gz8 = {0, 0, 0, 0, 0, 0, 0, 0};
      __builtin_amdgcn_tensor_load_to_lds(g0, g1, gz4, gz4, gz8, 0);
      __builtin_amdgcn_s_wait_tensorcnt((short)0);
    }
    __syncthreads();

    // ---- compute: 16 K-steps of 4 WMMAs from LDS fragments ----
    const int ar = 16 * w + l;
#pragma unroll
    for (int kk = 0; kk < BK; kk += 4) {
      const int kc = kk + 2 * half;
      const v2f a  = *(const v2f*)&s_X[ar][kc];
      const v2f b0 = *(const v2f*)&s_W[ 0 + l][kc];
      const v2f b1 = *(const v2f*)&s_W[16 + l][kc];
      const v2f b2 = *(const v2f*)&s_W[32 + l][kc];
      const v2f b3 = *(const v2f*)&s_W[48 + l][kc];
      acc0 = __builtin_amdgcn_wmma_f32_16x16x4_f32(false, a, false, b0,
                                                   (short)0, acc0, false, false);
      acc1 = __builtin_amdgcn_wmma_f32_16x16x4_f32(false, a, false, b1,
                                                   (short)0, acc1, false, false);
      acc2 = __builtin_amdgcn_wmma_f32_16x16x4_f32(false, a, false, b2,
                                                   (short)0, acc2, false, false);
      acc3 = __builtin_amdgcn_wmma_f32_16x16x4_f32(false, a, false, b3,
                                                   (short)0, acc3, false, false);
    }
    __syncthreads();
  }

  // ---- store: D layout lane<16 -> (M=i, N=lane); lane>=16 -> (M=8+i) ----
  const int mb = m0 + 16 * w;
  const int Bb = store_tb ? (M / Tdim) : 0;
  v8f accs[4] = {acc0, acc1, acc2, acc3};
#pragma unroll
  for (int j = 0; j < 4; ++j) {
    const int   n  = n0 + 16 * j + l;
    const float bn2 = bias ? bias[n] : 0.0f;
#pragma unroll
    for (int i = 0; i < 8; ++i) {
      const int   m   = mb + half * 8 + i;
      const float val = accs[j][i] + bn2;
      if (store_tb) {
        Y[((size_t)(m % Tdim) * Bb + (m / Tdim)) * N + n] = val;   // (T,B,N)
      } else {
        Y[(size_t)m * N + n] = val;                                // (M,N)
      }
    }
  }
}

// ---------------------------------------------------------------------------
// Fused score -> mask -> softmax -> context kernel.
// One 256-thread block (8 waves) per (b,t). Thread k owns knowledge slot k:
//   score_k = sum_e v[e] * tanh(wq[bt,e] + uh[b,k,e])     (512-deep, float4)
// then a block softmax over k (length-masked), then the block cooperatively
// forms c[bt,:] = align · know_bank[b] and fills the concat buffer
// [c | target | ctx] for the final WMMA projection.
// ---------------------------------------------------------------------------
__global__ void score_softmax_ctx(const float* __restrict__ wq,
                                  const float* __restrict__ uh,
                                  const float* __restrict__ v,
                                  const int*   __restrict__ know_len,
                                  const float* __restrict__ know_bank,
                                  const float* __restrict__ target,
                                  const float* __restrict__ ctxTB,
                                  float* __restrict__ align_out,
                                  float* __restrict__ c_out,
                                  float* __restrict__ concat)
{
  const int bt  = blockIdx.x;
  const int b   = bt / T_;
  const int t   = bt - b * T_;
  const int tid = threadIdx.x;

  __shared__ float s_wq[D_];
  __shared__ float s_v [D_];
  __shared__ float s_al[K_];
  __shared__ float red[8];

  for (int e = tid; e < D_; e += 256) {
    s_wq[e] = wq[(size_t)bt * D_ + e];
    s_v [e] = v[e];
  }
  __syncthreads();

  const int len = know_len[b];
  const int k   = tid;
  const float* __restrict__ uhrow = uh + ((size_t)b * K_ + k) * D_;

  float sc = 0.0f;
  for (int e = 0; e < D_; e += 4) {
    const float4 u = *(const float4*)(uhrow + e);
    sc += s_v[e + 0] * tanhf(s_wq[e + 0] + u.x);
    sc += s_v[e + 1] * tanhf(s_wq[e + 1] + u.y);
    sc += s_v[e + 2] * tanhf(s_wq[e + 2] + u.z);
    sc += s_v[e + 3] * tanhf(s_wq[e + 3] + u.w);
  }
  if (k >= len) sc = -INFINITY;             // length mask (len >= 1 always)

  // ---- block max (wave32 shuffle + cross-wave LDS) ----
  float m = sc;
  for (int off = 16; off > 0; off >>= 1) m = fmaxf(m, __shfl_xor(m, off, 32));
  if ((tid & 31) == 0) red[tid >> 5] = m;
  __syncthreads();
  if (tid == 0) {
    float mm = red[0];
    for (int i = 1; i < 8; ++i) mm = fmaxf(mm, red[i]);
    red[0] = mm;
  }
  __syncthreads();
  const float mx = red[0];
  __syncthreads();                           // red reused below

  // ---- exp + block sum ----
  const float ex = __expf(sc - mx);          // masked rows -> exp(-inf) = 0
  float s = ex;
  for (int off = 16; off > 0; off >>= 1) s += __shfl_xor(s, off, 32);
  if ((tid & 31) == 0) red[tid >> 5] = s;
  __syncthreads();
  if (tid == 0) {
    float ss = red[0];
    for (int i = 1; i < 8; ++i) ss += red[i];
    red[0] = ss;
  }
  __syncthreads();
  const float al = ex / red[0];              // len>=1 -> sum > 0, no NaN

  align_out[((size_t)t * B_ + b) * K_ + k] = al;   // (T,B,K)
  s_al[k] = al;
  __syncthreads();

  // ---- c = align · know_bank[b]  +  build concat row ----
  for (int d = tid; d < D_; d += 256) {
    float acc = 0.0f;
    const float* __restrict__ kb = know_bank + (size_t)b * K_ * D_ + d;
#pragma unroll 4
    for (int kk = 0; kk < K_; ++kk) acc += s_al[kk] * kb[(size_t)kk * D_];
    c_out [((size_t)t * B_ + b) * D_ + d]   = acc;                       // (T,B,D)
    concat[(size_t)bt * F_ + d]             = acc;                       // c
    concat[(size_t)bt * F_ + D_ + d]        = target[(size_t)bt * D_ + d];
    concat[(size_t)bt * F_ + 2 * D_ + d]    = ctxTB[((size_t)t * B_ + b) * D_ + d];
  }
}

// ---------------------------------------------------------------------------
extern "C" void kernel_launch(void* const* d_in, const int* in_sizes, int n_in,
                              void* d_out, int out_size, void* d_ws, size_t ws_size,
                              hipStream_t stream)
{
  (void)in_sizes; (void)n_in; (void)out_size; (void)ws_size;

  const float* target    = (const float*)d_in[0];   // (B,T,D)
  const float* know_bank = (const float*)d_in[1];   // (B,K,D)
  const float* ctxTB     = (const float*)d_in[2];   // (T,B,D)
  const int*   know_len  = (const int*)  d_in[3];   // (B,)
  const float* W_know    = (const float*)d_in[4];   // (D,D)
  const float* W_query   = (const float*)d_in[5];   // (D,D)
  const float* b_query   = (const float*)d_in[6];   // (D,)
  const float* v         = (const float*)d_in[7];   // (D,)
  const float* W_out     = (const float*)d_in[8];   // (D,F)
  const float* b_out     = (const float*)d_in[9];   // (D,)

  // d_out = [attn (T,B,D) | align (T,B,K) | c (T,B,D)] flat
  float* attn_out  = (float*)d_out;
  float* align_out = attn_out  + (size_t)T_ * B_ * D_;
  float* c_out     = align_out + (size_t)T_ * B_ * K_;

  // workspace: wq (B*T,D) | uh (B*K,D) | concat (B*T,F)   ~12.6 MB total
  char*  ws     = (char*)d_ws;
  float* wq     = (float*)ws;
  float* uh     = (float*)(ws + (size_t)B_ * T_ * D_ * sizeof(float));
  float* concat = (float*)(ws + ((size_t)B_ * T_ * D_ + (size_t)B_ * K_ * D_) * sizeof(float));

  auto launch_gemm = [&](const float* X, const float* W, const float* bias,
                         float* Y, int M, int N, int K, int tb) {
    const int grid = (M / BM) * (N / BN);   // one 128x64 tile per block
    hipLaunchKernelGGL(wmma_gemm_xwt, dim3(grid), dim3(256), 0, stream,
                       X, W, bias, Y, M, N, K, T_, tb);
  };

  // 1) wq = target·W_qᵀ + b_q   and   uh = know_bank·W_kᵀ
  launch_gemm(target,    W_query, b_query, wq, B_ * T_, D_, D_, 0);
  launch_gemm(know_bank, W_know,  nullptr, uh, B_ * K_, D_, D_, 0);

  // 2) fused score / softmax / context / concat-build
  hipLaunchKernelGGL(score_softmax_ctx, dim3(B_ * T_), dim3(256), 0, stream,
                     wq, uh, v, know_len, know_bank, target, ctxTB,
                     align_out, c_out, concat);

  // 3) attn = concat·W_outᵀ + b_out, stored directly as (T,B,D)
  launch_gemm(concat, W_out, b_out, attn_out, B_ * T_, D_, F_, 1);
}